// Rocket_75067438399920
// MI455X (gfx1250) — compile-verified
//
#include <hip/hip_runtime.h>
#include <math.h>

// ---------------------------------------------------------------------------
// ROCKET-256 fused feature extractor for MI455X (gfx1250, wave32, WMMA).
//   Stage 1: per-(b,c) mean/rstd (1280 rows).
//   Stage 2: device-side setup: cumsums, (l,d,p) grouping (<=16 kernels per
//            group -> WMMA M dim), dense zero-padded A_dense[256][112].
//   Stage 3: per (group, batch) wave32 block:
//            - A fragments pre-scaled by rstd (normalization folded into A)
//            - mean-correction + bias folded into the WMMA C seed
//            - time tiles double-buffered (ping-pong B gather prefetch
//              overlaps the serially-dependent v_wmma_f32_16x16x4_f32 chain)
//            - fused PPV + online softmax-max (exp2-based LSE).
// ---------------------------------------------------------------------------

typedef float v2f __attribute__((ext_vector_type(2)));
typedef float v8f __attribute__((ext_vector_type(8)));

#define RB   128
#define RC   10
#define RL   16384
#define NK   256
#define ASTR 112            // padded K stride (max K = 10*11 = 110)
#define T2   1442.69504f    // 1000 * log2(e): softmax temp folded into exp2

// -------------------------- Stage 1: row stats ------------------------------
__global__ void rocket_stats_kernel(const float* __restrict__ x,
                                    float* __restrict__ mean,
                                    float* __restrict__ rstd) {
    const int row = blockIdx.x;                 // b*RC + c, 0..1279
    const float* xr = x + (size_t)row * RL;
    float s = 0.f, ss = 0.f;
    for (int i = threadIdx.x; i < RL; i += 256) {
        float v = xr[i];
        s += v; ss += v * v;
    }
    __shared__ float sh1[256], sh2[256];
    sh1[threadIdx.x] = s; sh2[threadIdx.x] = ss;
    __syncthreads();
    for (int o = 128; o > 0; o >>= 1) {
        if (threadIdx.x < o) {
            sh1[threadIdx.x] += sh1[threadIdx.x + o];
            sh2[threadIdx.x] += sh2[threadIdx.x + o];
        }
        __syncthreads();
    }
    if (threadIdx.x == 0) {
        float mu  = sh1[0] * (1.f / (float)RL);
        float var = (sh2[0] - sh1[0] * mu) * (1.f / ((float)RL - 1.f));
        var = fmaxf(var, 0.f);
        mean[row] = mu;
        rstd[row] = 1.f / (sqrtf(var) + 1e-10f);
    }
}

// -------------------------- Stage 2: device setup ---------------------------
__global__ void rocket_setup_kernel(const float* __restrict__ weights,
                                    const int* __restrict__ lengths,
                                    const int* __restrict__ dils,
                                    const int* __restrict__ pads,
                                    const int* __restrict__ nci,
                                    const int* __restrict__ cidx,
                                    int* __restrict__ wstart,
                                    int* __restrict__ cstart,
                                    int* __restrict__ g_len,
                                    int* __restrict__ g_dil,
                                    int* __restrict__ g_pad,
                                    int* __restrict__ g_cnt,
                                    int* __restrict__ g_lout,
                                    int* __restrict__ g_kid,
                                    float* __restrict__ A_dense) {
    const int tid = threadIdx.x;
    if (tid == 0) {
        int wacc = 0, cacc = 0;
        for (int k = 0; k < NK; ++k) {
            wstart[k] = wacc; cstart[k] = cacc;
            wacc += lengths[k] * nci[k];
            cacc += nci[k];
        }
        for (int g = 0; g < NK; ++g) g_cnt[g] = 0;
        int ng = 0;
        for (int k = 0; k < NK; ++k) {
            int l = lengths[k], d = dils[k], p = pads[k];
            int tgt = -1;
            for (int g = 0; g < ng; ++g) {
                if (g_len[g] == l && g_dil[g] == d && g_pad[g] == p &&
                    g_cnt[g] < 16) { tgt = g; break; }
            }
            if (tgt < 0) {
                tgt = ng++;
                g_len[tgt] = l; g_dil[tgt] = d; g_pad[tgt] = p;
                g_lout[tgt] = RL + 2 * p - (l - 1) * d;
            }
            g_kid[tgt * 16 + g_cnt[tgt]] = k;
            g_cnt[tgt]++;
        }
    }
    __syncthreads();
    const int k = tid;
    float* Ak = A_dense + k * ASTR;
    for (int r = 0; r < ASTR; ++r) Ak[r] = 0.f;
    const int l  = lengths[k];
    const int nc = nci[k];
    const int wst = wstart[k];
    const int cst = cstart[k];
    for (int ci = 0; ci < nc; ++ci) {
        int c = cidx[cst + ci];
        for (int j = 0; j < l; ++j)
            Ak[c * l + j] = weights[wst + ci * l + j];
    }
}

// -------------------------- Stage 3: WMMA conv + features -------------------
// Gather one 16-time B tile into register array BV (static indexing only).
#define ROCKET_GATHER(BV, T0, T0N)                                           \
    do {                                                                     \
        if ((T0) >= p && (T0) <= t0hi) {                                     \
            _Pragma("unroll")                                                \
            for (int c = 0; c < KC; ++c) {                                   \
                BV[c].x = xb[(T0N) + foffx[c]];                              \
                BV[c].y = xb[(T0N) + foffy[c]];                              \
            }                                                                \
        } else {                                                             \
            _Pragma("unroll")                                                \
            for (int c = 0; c < KC; ++c) {                                   \
                _Pragma("unroll")                                            \
                for (int i = 0; i < 2; ++i) {                                \
                    int rc_ = min(4 * c + kk + i, K - 1);                    \
                    int ch  = rc_ / LEN;                                     \
                    int j   = rc_ - ch * LEN;                                \
                    int tt  = (T0N) + j * d - p;                             \
                    bool inb = ((unsigned)tt < (unsigned)RL);                \
                    int  ttc = inb ? tt : 0;                                 \
                    float xv = xb[ch * RL + ttc];                            \
                    float val = inb ? xv : mb[ch];                           \
                    if (i == 0) BV[c].x = val; else BV[c].y = val;           \
                }                                                            \
            }                                                                \
        }                                                                    \
    } while (0)

// WMMA chain (C seeded with bias-corr) + fused feature update at t = T0N.
#define ROCKET_COMPUTE(BV, T0N)                                              \
    do {                                                                     \
        v8f cfr = __builtin_amdgcn_wmma_f32_16x16x4_f32(                     \
            false, a_frag[0], false, BV[0], (short)0, effbv, false, false);  \
        _Pragma("unroll")                                                    \
        for (int c = 1; c < KC; ++c) {                                       \
            cfr = __builtin_amdgcn_wmma_f32_16x16x4_f32(                     \
                false, a_frag[c], false, BV[c], (short)0, cfr, false, false);\
        }                                                                    \
        const bool tin = ((T0N) < Lout);                                     \
        _Pragma("unroll")                                                    \
        for (int v = 0; v < 8; ++v) {                                        \
            float ym = tin ? cfr[v] : -1e30f;                                \
            acc_cnt[v] += (ym > 0.f) ? 1.f : 0.f;                            \
            float nm = fmaxf(acc_m[v], ym);                                  \
            float e1 = exp2f(T2 * (acc_m[v] - nm));                          \
            float e2 = exp2f(T2 * (ym - nm));                                \
            acc_s[v] = acc_s[v] * e1 + e2;                                   \
            acc_v[v] = acc_v[v] * e1 + e2 * ym;                              \
            acc_m[v] = nm;                                                   \
        }                                                                    \
    } while (0)

template <int LEN>
__global__ void __launch_bounds__(32)
rocket_conv_kernel(const float* __restrict__ x,
                   const float* __restrict__ mean,
                   const float* __restrict__ rstd,
                   const float* __restrict__ A_dense,
                   const float* __restrict__ biases,
                   const int* __restrict__ g_len,
                   const int* __restrict__ g_dil,
                   const int* __restrict__ g_pad,
                   const int* __restrict__ g_cnt,
                   const int* __restrict__ g_lout,
                   const int* __restrict__ g_kid,
                   float* __restrict__ out) {
    const int g = blockIdx.x;
    const int b = blockIdx.y;
    const int cnt = g_cnt[g];
    if (cnt == 0 || g_len[g] != LEN) return;   // block-uniform exit

    constexpr int K  = RC * LEN;               // 70 / 90 / 110
    constexpr int KC = (K + 3) / 4;            // 18 / 23 / 28 k-chunks

    const int d    = g_dil[g];
    const int p    = g_pad[g];
    const int Lout = g_lout[g];

    const int lane = threadIdx.x;              // wave32
    const int n    = lane & 15;                // N index (time within tile)
    const int hi   = lane >> 4;
    const int kk   = hi * 2;                   // K sub-offset per ISA layout

    const float* xb = x + (size_t)b * RC * RL;
    const float* mb = mean + b * RC;
    const float* rb = rstd + b * RC;

    // --- A fragments, pre-scaled by rstd; flat gather offsets hoisted.
    const int  m     = n;
    const bool mval  = (m < cnt);
    const int  kid_m = mval ? g_kid[g * 16 + m] : 0;
    v2f a_frag[KC];
    int foffx[KC], foffy[KC];
#pragma unroll
    for (int c = 0; c < KC; ++c) {
        int r0  = 4 * c + kk;
        int rc0 = min(r0, K - 1), rc1 = min(r0 + 1, K - 1);
        int ch0 = rc0 / LEN, j0 = rc0 - ch0 * LEN;
        int ch1 = rc1 / LEN, j1 = rc1 - ch1 * LEN;
        float ax = mval ? A_dense[kid_m * ASTR + r0]     * rb[ch0] : 0.f;
        float ay = mval ? A_dense[kid_m * ASTR + r0 + 1] * rb[ch1] : 0.f;
        a_frag[c].x = ax; a_frag[c].y = ay;
        foffx[c] = ch0 * RL + j0 * d - p;
        foffy[c] = ch1 * RL + j1 * d - p;
    }

    // --- mean correction via one WMMA chain: corr[m] = sum_r a[m,r]*mean[ch]
    v8f corrf = {};
#pragma unroll
    for (int c = 0; c < KC; ++c) {
        int r0  = 4 * c + kk;
        int rc0 = min(r0, K - 1), rc1 = min(r0 + 1, K - 1);
        v2f bm;
        bm.x = mb[rc0 / LEN];
        bm.y = mb[rc1 / LEN];
        corrf = __builtin_amdgcn_wmma_f32_16x16x4_f32(
            false, a_frag[c], false, bm, (short)0, corrf, false, false);
    }

    // --- per-lane feature rows follow the C/D layout: row = v + 8*hi
    int kidf[8]; bool valf[8];
    v8f effbv;                                  // bias - corr, as WMMA C seed
#pragma unroll
    for (int v = 0; v < 8; ++v) {
        int row  = v + 8 * hi;
        valf[v]  = (row < cnt);
        kidf[v]  = valf[v] ? g_kid[g * 16 + row] : 0;
        effbv[v] = biases[kidf[v]] - corrf[v];
    }
    float acc_cnt[8], acc_m[8], acc_s[8], acc_v[8];
#pragma unroll
    for (int v = 0; v < 8; ++v) {
        acc_cnt[v] = 0.f; acc_m[v] = -1e30f; acc_s[v] = 0.f; acc_v[v] = 0.f;
    }

    // tiles with t0 in [p, t0hi] cannot touch out-of-range x for any tap
    const int t0hi  = RL - 16 - (LEN - 1) * d + p;
    const int ntile = (Lout + 15) >> 4;

    // --- double-buffered tile pipeline: prefetch next tile's B while the
    //     current WMMA chain runs. Overshoot tiles are gathered from clamped
    //     (safe) addresses; their features self-mask via t0n >= Lout.
    v2f bvA[KC], bvB[KC];
    ROCKET_GATHER(bvA, 0, n);
    for (int tile = 0; tile < ntile; tile += 2) {
        {   // prefetch tile+1 into bvB
            int tc = min(tile + 1, ntile - 1);
            int pt0 = tc * 16, pt0n = pt0 + n;
            ROCKET_GATHER(bvB, pt0, pt0n);
        }
        {   // compute tile from bvA
            int t0n = tile * 16 + n;
            ROCKET_COMPUTE(bvA, t0n);
        }
        {   // prefetch tile+2 into bvA
            int tc = min(tile + 2, ntile - 1);
            int pt0 = tc * 16, pt0n = pt0 + n;
            ROCKET_GATHER(bvA, pt0, pt0n);
        }
        {   // compute tile+1 from bvB (auto-masked if past the end)
            int t0n = (tile + 1) * 16 + n;
            ROCKET_COMPUTE(bvB, t0n);
        }
    }

    // --- reduce over the 16 time-lanes sharing each row (log-sum-exp merge)
#pragma unroll
    for (int v = 0; v < 8; ++v) {
        float c_ = acc_cnt[v], m_ = acc_m[v], s_ = acc_s[v], w_ = acc_v[v];
#pragma unroll
        for (int msk = 1; msk < 16; msk <<= 1) {
            float om = __shfl_xor(m_, msk);
            float os = __shfl_xor(s_, msk);
            float ow = __shfl_xor(w_, msk);
            float oc = __shfl_xor(c_, msk);
            float nm = fmaxf(m_, om);
            float e1 = exp2f(T2 * (m_ - nm));
            float e2 = exp2f(T2 * (om - nm));
            s_ = s_ * e1 + os * e2;
            w_ = w_ * e1 + ow * e2;
            m_ = nm;
            c_ += oc;
        }
        if (n == 0 && valf[v]) {
            out[b * (2 * NK) + 2 * kidf[v] + 0] = c_ / (float)Lout;
            out[b * (2 * NK) + 2 * kidf[v] + 1] = w_ / s_;
        }
    }
}

// ------------------------------- launcher -----------------------------------
extern "C" void kernel_launch(void* const* d_in, const int* in_sizes, int n_in,
                              void* d_out, int out_size, void* d_ws, size_t ws_size,
                              hipStream_t stream) {
    const float* x       = (const float*)d_in[0];
    const float* weights = (const float*)d_in[1];
    const float* biases  = (const float*)d_in[2];
    const int*   lengths = (const int*)d_in[3];
    const int*   dils    = (const int*)d_in[4];
    const int*   pads    = (const int*)d_in[5];
    const int*   nci     = (const int*)d_in[6];
    const int*   cidx    = (const int*)d_in[7];
    float*       out     = (float*)d_out;

    float* mean    = (float*)d_ws;             // 1280 f32
    float* rstd    = mean + RB * RC;           // 1280 f32
    float* A_dense = rstd + RB * RC;           // 256*112 f32
    int*   ip      = (int*)(A_dense + NK * ASTR);
    int* wstart = ip;            ip += NK;
    int* cstart = ip;            ip += NK;
    int* g_len  = ip;            ip += NK;
    int* g_dil  = ip;            ip += NK;
    int* g_pad  = ip;            ip += NK;
    int* g_cnt  = ip;            ip += NK;
    int* g_lout = ip;            ip += NK;
    int* g_kid  = ip;            ip += NK * 16;

    rocket_stats_kernel<<<RB * RC, 256, 0, stream>>>(x, mean, rstd);
    rocket_setup_kernel<<<1, 256, 0, stream>>>(weights, lengths, dils, pads,
                                               nci, cidx, wstart, cstart,
                                               g_len, g_dil, g_pad, g_cnt,
                                               g_lout, g_kid, A_dense);
    dim3 grid(NK, RB);
    rocket_conv_kernel<7><<<grid, 32, 0, stream>>>(x, mean, rstd, A_dense, biases,
        g_len, g_dil, g_pad, g_cnt, g_lout, g_kid, out);
    rocket_conv_kernel<9><<<grid, 32, 0, stream>>>(x, mean, rstd, A_dense, biases,
        g_len, g_dil, g_pad, g_cnt, g_lout, g_kid, out);
    rocket_conv_kernel<11><<<grid, 32, 0, stream>>>(x, mean, rstd, A_dense, biases,
        g_len, g_dil, g_pad, g_cnt, g_lout, g_kid, out);
}